// StructuralConstraints_30897994727575
// MI455X (gfx1250) — compile-verified
//
#include <hip/hip_runtime.h>

// B=16, L=1024. Outputs: mask (16M f32) then updated (16M f32), concatenated.
#define LSEQ 1024
#define NB   16
#define ROWS 8
#define NTHREADS 256

typedef float v4f __attribute__((ext_vector_type(4)));

#if defined(__has_builtin)
#  if __has_builtin(__builtin_amdgcn_global_load_async_to_lds_b32) && \
      __has_builtin(__builtin_amdgcn_s_wait_asynccnt)
#    define USE_ASYNC_LDS 1
#  endif
#endif
#ifndef USE_ASYNC_LDS
#  define USE_ASYNC_LDS 0
#endif

#if USE_ASYNC_LDS
typedef __attribute__((address_space(1))) int g_int;
typedef __attribute__((address_space(3))) int l_int;
#endif

__global__ __launch_bounds__(NTHREADS)
void rna_struct_kernel(const float* __restrict__ bp,
                       const int*   __restrict__ seq,
                       const int*   __restrict__ slen,
                       float*       __restrict__ mask_out,
                       float*       __restrict__ upd_out)
{
    __shared__ int   sSeq[LSEQ];
    __shared__ int   sPKa[LSEQ];    // a0'(9b) | smx(5b)<<10 | a1'(9b)<<16
    __shared__ int   sPKb[LSEQ];    // a2'(9b) | a3'(9b)<<16          (sentinel 256)
    __shared__ float sE[1024];      // [0..255] = E table, [256..1023] = 0
    __shared__ float sV96[96];      // [0..15] = VALID, [16..95] = 0

    const int t   = threadIdx.x;
    const int b   = blockIdx.y;
    const int i0  = blockIdx.x * ROWS;
    const int len = slen[b];

    // ---------------- stage sequences[b,:] into LDS (ASYNCcnt path) ----------------
    const int* seqb = seq + b * LSEQ;
#if USE_ASYNC_LDS
    {
        #pragma unroll
        for (int q = 0; q < LSEQ / NTHREADS; ++q) {
            const int k = t + q * NTHREADS;
            __builtin_amdgcn_global_load_async_to_lds_b32(
                (g_int*)(const_cast<int*>(seqb) + k),
                (l_int*)(&sSeq[k]),
                0, 0);
        }
        __builtin_amdgcn_s_wait_asynccnt(0);
    }
#else
    {
        #pragma unroll
        for (int q = 0; q < LSEQ / NTHREADS; ++q) {
            const int k = t + q * NTHREADS;
            sSeq[k] = seqb[k];
        }
    }
#endif

    // ---------------- build E (1024, zero-padded) and VALID (96) tables ----------------
    {
        const unsigned vm = 0x4A52u;           // bit (a*4+b) set iff pair valid
        if (t < 96) sV96[t] = (t < 16 && ((vm >> t) & 1u)) ? 1.0f : 0.0f;
        const int ab = t >> 4, cd = t & 15;
        float val = 0.0f;
        if (((vm >> ab) & 1u) && ((vm >> cd) & 1u)) {
            switch (t) {                        // flat index a*64+b*16+c*4+d
                case 17:  val = 0.9f; break;    // (A,U,A,U)
                case 25:  val = 0.8f; break;    // (A,U,G,U)
                case 27:  val = 1.1f; break;    // (A,U,G,C)
                case 145: val = 0.8f; break;    // (G,U,A,U)
                case 153: val = 0.7f; break;    // (G,U,G,U)
                case 155: val = 1.0f; break;    // (G,U,G,C)
                case 177: val = 1.1f; break;    // (G,C,A,U)
                case 185: val = 1.0f; break;    // (G,C,G,U)
                case 187: val = 1.3f; break;    // (G,C,G,C)
                default:  val = 0.5f; break;
            }
        }
        sE[t]       = val;
        sE[t + 256] = 0.0f;
        sE[t + 512] = 0.0f;
        sE[t + 768] = 0.0f;
    }
    __syncthreads();

    // ------- packed separable E-index parts, length gates baked in as sentinels -------
    #pragma unroll
    for (int q = 0; q < LSEQ / NTHREADS; ++q) {
        const int k  = t + q * NTHREADS;
        const int s  = sSeq[k];
        const int sm = sSeq[(k - 1) & (LSEQ - 1)];
        const int sp = sSeq[(k + 1) & (LSEQ - 1)];
        int a0 = s * 64 + sp * 4;   // row part of S(k, .)      gate: k < len-1
        int a1 = s * 16 + sm;       // col part of S(., k)      gate: k < len
        int a2 = sm * 64 + s * 4;   // row part of S(k-1, .+1)  gate: 1 <= k < len
        int a3 = sp * 16 + s;       // col part of S(.-1, k+1)  gate: k < len-1
        int sx = s;                 // gated code for VALID     gate: k < len
        if (k >= len - 1)           a0 = 256;
        if (k >= len)             { a1 = 256; sx = 16; }
        if (k < 1 || k >= len)      a2 = 256;
        if (k >= len - 1)           a3 = 256;
        sPKa[k] = a0 | (sx << 10) | (a1 << 16);
        sPKb[k] = a2 | (a3 << 16);
    }
    __syncthreads();

    // ---------------- per-thread column-fixed data (extract once) ----------------
    const int j0 = t * 4;
    int a0j[4], a1j[4], a2j[4], a3j[4], sjm[4];
    #pragma unroll
    for (int q = 0; q < 4; ++q) {
        const int j  = j0 + q;
        const int pa = sPKa[j];
        const int pb = sPKb[j];
        a0j[q] = pa & 0x3FF;
        sjm[q] = (pa >> 10) & 31;   // gated s_j (16 -> zero region of sV96)
        a1j[q] = pa >> 16;
        a2j[q] = pb & 0x3FF;
        a3j[q] = pb >> 16;
    }

    const size_t base = ((size_t)b << 20) + ((size_t)i0 << 10) + (size_t)j0;

    #pragma unroll
    for (int r = 0; r < ROWS; ++r) {
        const int    i   = i0 + r;
        const size_t off = base + ((size_t)r << 10);

        v4f bpv = __builtin_nontemporal_load((const v4f*)(bp + off));
        if (r < ROWS - 2)
            __builtin_prefetch(bp + off + 2 * LSEQ, 0, 3);   // global_prefetch_b8

        const int pa  = sPKa[i];
        const int pb  = sPKb[i];
        const int a0i = pa & 0x3FF;
        const int six = (pa >> 8) & 124;   // ((pa>>10)&31)<<2 : gated 4*s_i (64 -> zeros)
        const int a1i = pa >> 16;
        const int a2i = pb & 0x3FF;
        const int a3i = pb >> 16;

        // VALID value (length gates baked in via sentinels), hoisted per row
        float vg[4];
        #pragma unroll
        for (int q = 0; q < 4; ++q)
            vg[q] = sV96[six + sjm[q]];

        float addv[4] = {0.0f, 0.0f, 0.0f, 0.0f};

        // ---- upper-triangle terms: possible iff max_q d = j0+3-i >= 2 ----
        if (j0 + 3 - i >= 2) {
            #pragma unroll
            for (int q = 0; q < 4; ++q) {
                const int d = j0 + q - i;
                const float e0 = sE[a0i + a1j[q]];   // S(i,j)     (len gates baked in)
                const float e1 = sE[a2i + a3j[q]];   // S(i-1,j+1)
                addv[q] = ((d >= 4) ? e0 : 0.0f) + ((d >= 2) ? e1 : 0.0f);
            }
        }
        // ---- lower-triangle terms: possible iff min_q d = j0-i <= -2 ----
        if (j0 - i <= -2) {
            #pragma unroll
            for (int q = 0; q < 4; ++q) {
                const int d = j0 + q - i;
                const float e2 = sE[a0j[q] + a1i];   // S(j,i)
                const float e3 = sE[a2j[q] + a3i];   // S(j-1,i+1)
                addv[q] += ((d <= -4) ? e2 : 0.0f) + ((d <= -2) ? e3 : 0.0f);
            }
        }

        v4f mv, uv;
        #pragma unroll
        for (int q = 0; q < 4; ++q) {
            const int d  = j0 + q - i;
            const int ad = d < 0 ? -d : d;
            mv[q] = (ad >= 3) ? vg[q] : 0.0f;
            uv[q] = bpv[q] + addv[q];
        }

        __builtin_nontemporal_store(mv, (v4f*)(mask_out + off));
        __builtin_nontemporal_store(uv, (v4f*)(upd_out  + off));
    }
}

extern "C" void kernel_launch(void* const* d_in, const int* in_sizes, int n_in,
                              void* d_out, int out_size, void* d_ws, size_t ws_size,
                              hipStream_t stream) {
    const float* bp   = (const float*)d_in[0];   // (16,1024,1024) f32
    const int*   seq  = (const int*)  d_in[1];   // (16,1024) i32
    const int*   slen = (const int*)  d_in[2];   // (16,) i32
    float* mask_out = (float*)d_out;
    float* upd_out  = mask_out + (size_t)NB * LSEQ * LSEQ;

    dim3 grid(LSEQ / ROWS, NB);
    rna_struct_kernel<<<grid, NTHREADS, 0, stream>>>(bp, seq, slen, mask_out, upd_out);
}